// PMLP_GCN_2216203125084
// MI455X (gfx1250) — compile-verified
//
#include <hip/hip_runtime.h>
#include <hip/hip_bf16.h>
#include <math.h>

// ---------------------------------------------------------------------------
// PMLP-GCN forward on MI455X (gfx1250).
//   layer: h = x @ W^T  (WMMA f32 16x16x4, exact fp32; W tile staged via TDM)
//          agg[dst] += h[src] * dinv[src]*dinv[dst]   (f32 global atomics, L2-resident)
//          y = relu((agg + b - mu) * rsqrt(var+eps))  (layers 1,2)
// ---------------------------------------------------------------------------

typedef __attribute__((ext_vector_type(2))) float v2f;
typedef __attribute__((ext_vector_type(8))) float v8f;
typedef __attribute__((ext_vector_type(4))) unsigned int v4u;
typedef __attribute__((ext_vector_type(8))) int v8i;
typedef __attribute__((ext_vector_type(4))) int v4i;

#define GK 256          // K is 256 for all three GEMMs
#define TILE_ROWS 128   // per block
#define TILE_COLS 64    // per block

// ---------------- degree / normalization prep (done once, reused 3x) -------

__global__ __launch_bounds__(256) void deg_kernel(const long long* __restrict__ dst,
                                                  float* __restrict__ deg, int E) {
    int t = blockIdx.x * 256 + threadIdx.x;
    if (t < E) unsafeAtomicAdd(&deg[(int)dst[t]], 1.0f);
}

__global__ __launch_bounds__(256) void dinv_kernel(const float* __restrict__ deg,
                                                   float* __restrict__ dinv, int n) {
    int t = blockIdx.x * 256 + threadIdx.x;
    if (t < n) {
        float d = deg[t];
        dinv[t] = (d > 0.0f) ? rsqrtf(d) : 0.0f;   // deg>=1 when >0, matches max(deg,1)
    }
}

__global__ __launch_bounds__(256) void edge_prep_kernel(const long long* __restrict__ src,
                                                        const long long* __restrict__ dst,
                                                        const float* __restrict__ dinv,
                                                        int* __restrict__ s32, int* __restrict__ d32,
                                                        float* __restrict__ wE, int E) {
    int t = blockIdx.x * 256 + threadIdx.x;
    if (t < E) {
        int s = (int)src[t], d = (int)dst[t];
        s32[t] = s; d32[t] = d;
        wE[t] = dinv[s] * dinv[d];
    }
}

// ---------------- GEMM: Y[N,M] = X[N,GK] @ W[M,GK]^T  (WMMA fp32) ----------

__global__ __launch_bounds__(256) void gemm_xwT_wmma(const float* __restrict__ X,
                                                     const float* __restrict__ W,
                                                     float* __restrict__ Y,
                                                     int nrows, int M) {
    __shared__ float Ws[TILE_COLS * GK];   // 64 KB weight tile

    const int tid = threadIdx.x;
    const int colBlock = blockIdx.y * TILE_COLS;

    // ---- Stage W tile (64 rows x 256 f32, contiguous) into LDS via the
    //      Tensor Data Mover: one 2D DMA issued by wave 0, instead of 4096
    //      vector loads.  D# layout per CDNA5 ISA ch.8 (group0/group1).
    if ((tid >> 5) == 0) {
        unsigned long long ga =
            (unsigned long long)(const void*)(W + (size_t)colBlock * GK);
        unsigned ldsOff = (unsigned)(size_t)Ws;   // low 32 bits = LDS byte offset
        v4u g0;
        g0[0] = 1u;                               // count=1, user mode, no gather
        g0[1] = ldsOff;                           // lds_addr (bytes)
        g0[2] = (unsigned)(ga & 0xFFFFFFFFu);     // global_addr[31:0]
        g0[3] = (unsigned)((ga >> 32) & 0x1FFFFFFu) | (2u << 30);  // addr[56:32] | type=2
        v8i g1;
        g1[0] = (int)(2u << 16);      // wg_mask=0 | data_size=2 (4 bytes)
        g1[1] = (int)(256u << 16);    // atomic_barrier_addr=0 | tensor_dim0.lo16=256
        g1[2] = (int)(64u << 16);     // tensor_dim0.hi16=0    | tensor_dim1.lo16=64
        g1[3] = (int)(256u << 16);    // tensor_dim1.hi16=0    | tile_dim0=256
        g1[4] = 64;                   // tile_dim1=64 | tile_dim2=0
        g1[5] = 256;                  // tensor_dim0_stride.lo32=256
        g1[6] = 0;                    // stride.hi16=0 | tensor_dim1_stride.lo16=0
        g1[7] = 0;                    // tensor_dim1_stride.hi32=0
        v4i gz4 = {0, 0, 0, 0};
        v8i gz8 = {0, 0, 0, 0, 0, 0, 0, 0};
        __builtin_amdgcn_tensor_load_to_lds(g0, g1, gz4, gz4, gz8, 0);
        __builtin_amdgcn_s_wait_tensorcnt(0);
    }
    __syncthreads();

    const int lane   = tid & 31;
    const int wave   = tid >> 5;
    const int waveRow = (wave >> 1) * 32;      // 0,32,64,96
    const int waveCol = (wave & 1) * 32;       // 0,32
    const int m      = lane & 15;
    const int khalf  = (lane >> 4) << 1;       // lanes 0-15 -> K{0,1}, 16-31 -> K{2,3}

    // Clamp out-of-range rows instead of predicating the loads: the clamped
    // rows only feed output rows >= nrows, which are never stored.  This keeps
    // the inner K-loop branch-free (no EXEC save/restore per iteration).
    const int r0 = blockIdx.x * TILE_ROWS + waveRow + m;
    const int r1 = r0 + 16;
    const int rc0 = r0 < nrows ? r0 : nrows - 1;
    const int rc1 = r1 < nrows ? r1 : nrows - 1;
    const float* x0 = X + (size_t)rc0 * GK;
    const float* x1 = X + (size_t)rc1 * GK;
    const float* w0 = Ws + (waveCol + m) * GK;
    const float* w1 = Ws + (waveCol + 16 + m) * GK;

    v8f c00 = {}, c01 = {}, c10 = {}, c11 = {};

#pragma unroll 4
    for (int k = 0; k < GK; k += 4) {
        const int kk = k + khalf;
        v2f a0 = *(const v2f*)(x0 + kk);                 // A: 16x4 fp32 fragment
        v2f a1 = *(const v2f*)(x1 + kk);
        v2f b0 = *(const v2f*)(w0 + kk);                 // B: 4x16 fp32 fragment (LDS)
        v2f b1 = *(const v2f*)(w1 + kk);
        c00 = __builtin_amdgcn_wmma_f32_16x16x4_f32(false, a0, false, b0, (short)0, c00, false, false);
        c01 = __builtin_amdgcn_wmma_f32_16x16x4_f32(false, a0, false, b1, (short)0, c01, false, false);
        c10 = __builtin_amdgcn_wmma_f32_16x16x4_f32(false, a1, false, b0, (short)0, c10, false, false);
        c11 = __builtin_amdgcn_wmma_f32_16x16x4_f32(false, a1, false, b1, (short)0, c11, false, false);
    }

    // C/D layout: VGPR i -> row i (lanes 0-15) / row i+8 (lanes 16-31), col = lane&15
    const int colT0 = colBlock + waveCol + (lane & 15);
    const int colT1 = colT0 + 16;
    const int rowHalf = (lane >> 4) << 3;
    const int rowT0 = blockIdx.x * TILE_ROWS + waveRow + rowHalf;
    const int rowT1 = rowT0 + 16;
#pragma unroll
    for (int i = 0; i < 8; ++i) {
        int ra = rowT0 + i, rb = rowT1 + i;
        if (ra < nrows) {
            Y[(size_t)ra * M + colT0] = c00[i];
            Y[(size_t)ra * M + colT1] = c01[i];
        }
        if (rb < nrows) {
            Y[(size_t)rb * M + colT0] = c10[i];
            Y[(size_t)rb * M + colT1] = c11[i];
        }
    }
}

// ---------------- edge aggregation: out[dst] += h[src] * w -----------------

__global__ __launch_bounds__(256) void agg_kernel(const float* __restrict__ h,
                                                  const int* __restrict__ src,
                                                  const int* __restrict__ dst,
                                                  const float* __restrict__ wE,
                                                  float* __restrict__ out,
                                                  int E, int lgC4, int C) {
    long long t = (long long)blockIdx.x * 256 + threadIdx.x;
    if (t >= ((long long)E << lgC4)) return;
    int e  = (int)(t >> lgC4);
    int cg = ((int)t) & ((1 << lgC4) - 1);
    int s = src[e], d = dst[e];
    float w = wE[e];
    float4 hv = *(const float4*)(h + (size_t)s * C + (cg << 2));
    float* op = out + (size_t)d * C + (cg << 2);
    unsafeAtomicAdd(op + 0, hv.x * w);
    unsafeAtomicAdd(op + 1, hv.y * w);
    unsafeAtomicAdd(op + 2, hv.z * w);
    unsafeAtomicAdd(op + 3, hv.w * w);
}

// ---------------- batchnorm (C=256, affine=False) + bias + relu ------------

__global__ __launch_bounds__(256) void bn_stats_kernel(const float* __restrict__ x,
                                                       const float* __restrict__ bias,
                                                       float* __restrict__ stats, int n) {
    int c = threadIdx.x;                       // one channel per thread, coalesced rows
    float b = bias[c];
    float s = 0.0f, q = 0.0f;
    for (int r = blockIdx.x; r < n; r += gridDim.x) {
        float v = x[(size_t)r * 256 + c] + b;
        s += v; q += v * v;
    }
    unsafeAtomicAdd(&stats[c], s);
    unsafeAtomicAdd(&stats[256 + c], q);
}

__global__ void bn_final_kernel(const float* __restrict__ stats,
                                float* __restrict__ musc, float n) {
    int c = threadIdx.x;
    float mu  = stats[c] / n;
    float var = stats[256 + c] / n - mu * mu;
    musc[c]       = mu;
    musc[256 + c] = rsqrtf(var + 1e-5f);
}

__global__ __launch_bounds__(256) void bn_norm_kernel(const float* __restrict__ x,
                                                      const float* __restrict__ bias,
                                                      const float* __restrict__ musc,
                                                      float* __restrict__ y, int n) {
    long long t = (long long)blockIdx.x * 256 + threadIdx.x;
    if (t >= (long long)n * 256) return;
    int c = (int)(t & 255);
    float v = (x[t] + bias[c] - musc[c]) * musc[256 + c];
    y[t] = v > 0.0f ? v : 0.0f;
}

__global__ __launch_bounds__(256) void bias_out_kernel(float* __restrict__ out,
                                                       const float* __restrict__ b,
                                                       long long total) {
    long long t = (long long)blockIdx.x * 256 + threadIdx.x;
    if (t < total) out[t] += b[(int)(t & 63)];
}

// ---------------------------------------------------------------------------

extern "C" void kernel_launch(void* const* d_in, const int* in_sizes, int n_in,
                              void* d_out, int out_size, void* d_ws, size_t ws_size,
                              hipStream_t stream) {
    const float*     X  = (const float*)d_in[0];
    const long long* EI = (const long long*)d_in[1];   // [2,E] int64: src then dst
    const float* W1 = (const float*)d_in[2];
    const float* B1 = (const float*)d_in[3];
    const float* W2 = (const float*)d_in[4];
    const float* B2 = (const float*)d_in[5];
    const float* W3 = (const float*)d_in[6];
    const float* B3 = (const float*)d_in[7];
    const int N = in_sizes[0] / 256;   // 100000
    const int E = in_sizes[1] / 2;     // 3200000
    float* out = (float*)d_out;

    // workspace carve
    char* p = (char*)d_ws;
    float* A    = (float*)p; p += (size_t)N * 256 * 4;   // ping
    float* Bf   = (float*)p; p += (size_t)N * 256 * 4;   // pong
    float* deg  = (float*)p; p += (size_t)N * 4;
    float* dinv = (float*)p; p += (size_t)N * 4;
    float* stats= (float*)p; p += 512 * 4;
    float* musc = (float*)p; p += 512 * 4;
    int*   s32  = (int*)p;   p += (size_t)E * 4;
    int*   d32  = (int*)p;   p += (size_t)E * 4;
    float* wE   = (float*)p; p += (size_t)E * 4;

    const dim3 blk(256);
    const int eBlocks  = (E + 255) / 256;
    const int nBlocks  = (N + 255) / 256;
    const int aggBlk256 = (int)(((long long)E * 64 + 255) / 256);  // C=256 -> C/4=64
    const int aggBlk64  = (int)(((long long)E * 16 + 255) / 256);  // C=64  -> C/4=16
    const dim3 gemmGrid256((N + TILE_ROWS - 1) / TILE_ROWS, 256 / TILE_COLS);
    const dim3 gemmGrid64 ((N + TILE_ROWS - 1) / TILE_ROWS, 64 / TILE_COLS);

    // --- graph prep (deterministic, recomputed every call) ---
    (void)hipMemsetAsync(deg, 0, (size_t)N * 4, stream);
    deg_kernel<<<eBlocks, blk, 0, stream>>>(EI + E, deg, E);
    dinv_kernel<<<nBlocks, blk, 0, stream>>>(deg, dinv, N);
    edge_prep_kernel<<<eBlocks, blk, 0, stream>>>(EI, EI + E, dinv, s32, d32, wE, E);

    // --- layer 1: X -> A (gemm) -> Bf (agg) -> A (bn+relu) ---
    gemm_xwT_wmma<<<gemmGrid256, blk, 0, stream>>>(X, W1, A, N, 256);
    (void)hipMemsetAsync(Bf, 0, (size_t)N * 256 * 4, stream);
    agg_kernel<<<aggBlk256, blk, 0, stream>>>(A, s32, d32, wE, Bf, E, 6, 256);
    (void)hipMemsetAsync(stats, 0, 512 * 4, stream);
    bn_stats_kernel<<<2048, blk, 0, stream>>>(Bf, B1, stats, N);
    bn_final_kernel<<<1, blk, 0, stream>>>(stats, musc, (float)N);
    bn_norm_kernel<<<(int)(((long long)N * 256 + 255) / 256), blk, 0, stream>>>(Bf, B1, musc, A, N);

    // --- layer 2: A -> Bf (gemm) -> A (agg) -> Bf (bn+relu) ---
    gemm_xwT_wmma<<<gemmGrid256, blk, 0, stream>>>(A, W2, Bf, N, 256);
    (void)hipMemsetAsync(A, 0, (size_t)N * 256 * 4, stream);
    agg_kernel<<<aggBlk256, blk, 0, stream>>>(Bf, s32, d32, wE, A, E, 6, 256);
    (void)hipMemsetAsync(stats, 0, 512 * 4, stream);
    bn_stats_kernel<<<2048, blk, 0, stream>>>(A, B2, stats, N);
    bn_final_kernel<<<1, blk, 0, stream>>>(stats, musc, (float)N);
    bn_norm_kernel<<<(int)(((long long)N * 256 + 255) / 256), blk, 0, stream>>>(A, B2, musc, Bf, N);

    // --- layer 3: Bf -> A (gemm, M=64) -> d_out (agg) ; + b3 ---
    gemm_xwT_wmma<<<gemmGrid64, blk, 0, stream>>>(Bf, W3, A, N, 64);
    (void)hipMemsetAsync(out, 0, (size_t)N * 64 * 4, stream);
    agg_kernel<<<aggBlk64, blk, 0, stream>>>(A, s32, d32, wE, out, E, 4, 64);
    bias_out_kernel<<<(int)(((long long)N * 64 + 255) / 256), blk, 0, stream>>>(out, B3, (long long)N * 64);
}